// NSSMetric_53317724012633
// MI455X (gfx1250) — compile-verified
//
#include <hip/hip_runtime.h>

typedef __attribute__((ext_vector_type(2))) float v2f;
typedef __attribute__((ext_vector_type(8))) float v8f;

constexpr int BATCH = 64;
constexpr int H = 512;
constexpr int W = 512;
constexpr int NPIX = H * W;          // 262144 per batch
constexpr int FIX = 12;
constexpr int SEGS = 32;             // segments (blocks) per batch
constexpr int TPB = 256;             // 8 wave32s
constexpr int PER_SEG = NPIX / SEGS; // 8192 floats per block
constexpr int VEC_ITERS = PER_SEG / (TPB * 4); // 8 x float4 per thread

// Wave32 sum reduction via V_WMMA_F32_16X16X4_F32.
// A fragment per lane = {x, 0}: per ISA layout, lanes 0-15 supply A[m][K=0],
// lanes 16-31 supply A[m][K=2] (K=1,3 slots are zero). B = all ones, which is
// layout-independent, so D[m][n] = x_m + x_{m+16} for every n. Summing the 8
// C VGPRs in a lane gives the half-M column sum; one shfl_xor(16) completes
// the full 32-lane sum (broadcast to all lanes).
__device__ __forceinline__ float wave_reduce_sum_wmma(float x) {
    v2f a; a.x = x;    a.y = 0.0f;
    v2f b; b.x = 1.0f; b.y = 1.0f;
    v8f c = {};
    c = __builtin_amdgcn_wmma_f32_16x16x4_f32(
            /*neg_a=*/false, a, /*neg_b=*/false, b,
            /*c_mod=*/(short)0, c, /*reuse_a=*/false, /*reuse_b=*/false);
    float t = c[0] + c[1] + c[2] + c[3] + c[4] + c[5] + c[6] + c[7];
    t += __shfl_xor(t, 16, 32);
    return t;
}

// Pass 1: streaming (sum, sumsq) partials. One block per (batch, segment).
__global__ void __launch_bounds__(TPB)
nss_partial_kernel(const float* __restrict__ pred, float2* __restrict__ partial) {
    const int blk = blockIdx.x;
    const int b   = blk / SEGS;
    const int seg = blk % SEGS;
    const float4* base =
        (const float4*)(pred + (size_t)b * NPIX + (size_t)seg * PER_SEG);
    const int tid = threadIdx.x;

    float s = 0.0f, q = 0.0f;
#pragma unroll
    for (int j = 0; j < VEC_ITERS; ++j) {
        float4 x = base[j * TPB + tid];           // coalesced b128 stream
        s += x.x; s += x.y; s += x.z; s += x.w;
        q = fmaf(x.x, x.x, q);
        q = fmaf(x.y, x.y, q);
        q = fmaf(x.z, x.z, q);
        q = fmaf(x.w, x.w, q);
    }

    // Wave-level reduction on the matrix pipe (EXEC is all-ones here:
    // full blocks, no divergence before this point).
    float wsum = wave_reduce_sum_wmma(s);
    float wsq  = wave_reduce_sum_wmma(q);

    __shared__ float sh[(TPB / 32) * 2];
    const int wave = tid >> 5;
    const int lane = tid & 31;
    if (lane == 0) { sh[wave * 2] = wsum; sh[wave * 2 + 1] = wsq; }
    __syncthreads();
    if (tid == 0) {
        float S = 0.0f, Q = 0.0f;
        for (int wv = 0; wv < TPB / 32; ++wv) { S += sh[wv * 2]; Q += sh[wv * 2 + 1]; }
        partial[blk] = make_float2(S, Q);         // deterministic, no atomics
    }
}

// Pass 2: one block, thread t owns batch t. Combine partials, compute
// mean / unbiased std, gather+normalize only the 12 fixation points
// (hits L2: 192 MB holds the whole 64 MB map), then batch mean.
__global__ void __launch_bounds__(64)
nss_final_kernel(const float* __restrict__ pred,
                 const float* __restrict__ coords,
                 const float2* __restrict__ partial,
                 float* __restrict__ out) {
    const int b = threadIdx.x;                    // 0..63
    float S = 0.0f, Q = 0.0f;
    for (int s = 0; s < SEGS; ++s) {              // fixed-order combine
        float2 p = partial[b * SEGS + s];
        S += p.x; Q += p.y;
    }
    const float nf   = (float)NPIX;
    const float mean = S / nf;
    float var = (Q - S * mean) / (nf - 1.0f);     // (Q - S^2/N) / (N-1), ddof=1
    var = fmaxf(var, 0.0f);
    const float inv = 1.0f / (sqrtf(var) + 1e-8f);

    float acc = 0.0f;
    int   cnt = 0;
    for (int f = 0; f < FIX; ++f) {
        const float cx = coords[(b * FIX + f) * 2 + 0];
        const float cy = coords[(b * FIX + f) * 2 + 1];
        const int px = (int)cx;                   // trunc toward zero
        const int py = (int)cy;
        const bool valid = (px >= 0) && (px < W) && (py >= 0) && (py < H);
        const int pxc = min(max(px, 0), W - 1);
        const int pyc = min(max(py, 0), H - 1);
        const float v = pred[(size_t)b * NPIX + (size_t)pyc * W + pxc];
        if (valid) { acc += (v - mean) * inv; ++cnt; }
    }
    const float per = (cnt > 0) ? acc / (float)cnt : 0.0f;

    __shared__ float sh[64];
    sh[b] = per;
    __syncthreads();
    if (b == 0) {
        float t = 0.0f;
        for (int i = 0; i < 64; ++i) t += sh[i];  // fixed order
        out[0] = t * (1.0f / 64.0f);
    }
}

extern "C" void kernel_launch(void* const* d_in, const int* in_sizes, int n_in,
                              void* d_out, int out_size, void* d_ws, size_t ws_size,
                              hipStream_t stream) {
    (void)in_sizes; (void)n_in; (void)out_size; (void)ws_size;
    const float* pred   = (const float*)d_in[0];   // (64,1,512,512) f32
    const float* coords = (const float*)d_in[1];   // (64,12,2) f32
    float*  out     = (float*)d_out;               // scalar f32
    float2* partial = (float2*)d_ws;               // 2048 * float2 = 16 KB scratch

    nss_partial_kernel<<<BATCH * SEGS, TPB, 0, stream>>>(pred, partial);
    nss_final_kernel<<<1, 64, 0, stream>>>(pred, coords, partial, out);
}